// TreeDecoder_20804821582284
// MI455X (gfx1250) — compile-verified
//
#include <hip/hip_runtime.h>

// TreeDecoder pipeline for MI455X (gfx1250, wave32).
// Phase 1: 4x GEMM+LeakyReLU via V_WMMA_F32_16X16X4_F32.
// Phase 2: 3x fused [BinaryTreeConv (gathered GEMM via WMMA) + TreeLayerNorm + LeakyReLU],
//          one workgroup per sample, accumulators in VGPRs, deterministic reduction.

typedef float v2f __attribute__((ext_vector_type(2)));
typedef float v8f __attribute__((ext_vector_type(8)));

#define SLOPE 0.01f
#define LN_EPS 1e-5f

__device__ __forceinline__ float leaky(float v) { return v > 0.0f ? v : SLOPE * v; }

__device__ __forceinline__ v8f wmma_f32(v2f a, v2f b, v8f c) {
  // D = A(16x4 f32) * B(4x16 f32) + C(16x16 f32)
  return __builtin_amdgcn_wmma_f32_16x16x4_f32(false, a, false, b, (short)0, c, false, false);
}

// ---------------------------------------------------------------------------
// GEMM + bias + LeakyReLU:  out[M,N] = leaky(X[M,K] @ W[N,K]^T + bias[N])
// One wave computes one 16x16 tile. 8 waves / block.
// ---------------------------------------------------------------------------
__global__ __launch_bounds__(256) void gemm_leaky_kernel(
    const float* __restrict__ X, const float* __restrict__ W,
    const float* __restrict__ bias, float* __restrict__ out,
    int M, int N, int K)
{
  const int lane = threadIdx.x & 31;
  const int wave = threadIdx.x >> 5;
  const int nl   = lane & 15;
  const int half = lane >> 4;       // 0: lanes 0-15 (K=0,1), 1: lanes 16-31 (K=2,3)
  const int koff = half * 2;

  const int tilesN = N >> 4;
  const int tilesM = M >> 4;
  int tile = blockIdx.x * 8 + wave;
  int mt = tile / tilesN;
  int nt = tile - mt * tilesN;
  if (mt >= tilesM) return;         // wave-uniform

  const float* arow = X + (long)(mt * 16 + nl) * K;   // A rows: M = lane&15
  const float* brow = W + (long)(nt * 16 + nl) * K;   // B cols: N = lane&15 (W row-major N,K)

  v8f acc = {0.f, 0.f, 0.f, 0.f, 0.f, 0.f, 0.f, 0.f};
  for (int kb = 0; kb < K; kb += 4) {
    v2f a = *(const v2f*)(arow + kb + koff);
    v2f b = *(const v2f*)(brow + kb + koff);
    acc = wmma_f32(a, b, acc);
  }

  const int ncol = nt * 16 + nl;
  const float bv = bias[ncol];
  long obase = (long)(mt * 16 + half * 8) * N + ncol;  // C/D: VGPR r -> row r + 8*half
#pragma unroll
  for (int r = 0; r < 8; ++r)
    out[obase + (long)r * N] = leaky(acc[r] + bv);
}

// ---------------------------------------------------------------------------
// Fused BinaryTreeConv + TreeLayerNorm + LeakyReLU.
// One workgroup (256 thr = 8 waves) per sample b.
//   conv: out[o, n+1] = sum_{c,k} cw[o,c,k] * x[b, c, idx[b, 3n+k]] + cb[o],  n=0..62
//   slot 0 = 0 before norm; layernorm over all O*64 elements (unbiased var, eps on std).
// GEMM view: A = cw (O x 3C), B[j][n] = x[b, j/3, idx[3n + j%3]]  (3C x 63, col 63 masked).
// ---------------------------------------------------------------------------
template <int C, int O>
__global__ __launch_bounds__(256) void conv_ln_leaky_kernel(
    const float* __restrict__ Xall,   // (B, C, 64)
    const int*   __restrict__ idxAll, // (B, 189)
    const float* __restrict__ Wc,     // (O, C, 3) == (O, 3C)
    const float* __restrict__ bc,     // (O,)
    float*       __restrict__ out)    // (B, O, 64)
{
  constexpr int K  = 3 * C;
  constexpr int MT = O / 16 / 8;      // m-tiles per wave (1, 2, 4)

  const int tid  = threadIdx.x;
  const int lane = tid & 31;
  const int wave = tid >> 5;
  const int nl   = lane & 15;
  const int half = lane >> 4;
  const int koff = half * 2;
  const int b    = blockIdx.x;

  __shared__ int   sidx[192];
  __shared__ float swsum[8], swsq[8];

  if (tid < 189) sidx[tid] = idxAll[b * 189 + tid];
  __syncthreads();

  const float* xb = Xall + (long)b * C * 64;

  v8f acc[MT][4];
#pragma unroll
  for (int m = 0; m < MT; ++m)
#pragma unroll
    for (int t = 0; t < 4; ++t)
      acc[m][t] = (v8f){0.f, 0.f, 0.f, 0.f, 0.f, 0.f, 0.f, 0.f};

  for (int kb = 0; kb < K; kb += 4) {
    const int j0 = kb + koff;
    // Gather B tiles for the 4 N-tiles (shared by all M-tiles of this wave)
    v2f bm[4];
#pragma unroll
    for (int t = 0; t < 4; ++t) {
      const int n  = t * 16 + nl;
      const int en = (n < 63) ? n : 0;   // col 63 is padding: safe read, masked later
#pragma unroll
      for (int e = 0; e < 2; ++e) {
        const int j = j0 + e;
        const int c = j / 3;
        const int k = j - c * 3;
        const int slot = sidx[en * 3 + k];
        bm[t][e] = xb[c * 64 + slot];
      }
    }
#pragma unroll
    for (int m = 0; m < MT; ++m) {
      const int mtile = wave * MT + m;
      v2f a = *(const v2f*)(Wc + (long)(mtile * 16 + nl) * K + j0);
#pragma unroll
      for (int t = 0; t < 4; ++t)
        acc[m][t] = wmma_f32(a, bm[t], acc[m][t]);
    }
  }

  // Add bias, accumulate masked sum / sumsq (zero slot-0 column contributes 0).
  float ls = 0.f, ls2 = 0.f;
#pragma unroll
  for (int m = 0; m < MT; ++m) {
    const int mtile = wave * MT + m;
#pragma unroll
    for (int t = 0; t < 4; ++t) {
      const int n = t * 16 + nl;
#pragma unroll
      for (int r = 0; r < 8; ++r) {
        const int o = mtile * 16 + half * 8 + r;
        float v = acc[m][t][r] + bc[o];
        acc[m][t][r] = v;
        if (n < 63) { ls += v; ls2 += v * v; }
      }
    }
  }

  // Deterministic reduction: shfl tree within wave, fixed-order combine across 8 waves.
#pragma unroll
  for (int off = 16; off > 0; off >>= 1) {
    ls  += __shfl_xor(ls,  off, 32);
    ls2 += __shfl_xor(ls2, off, 32);
  }
  if (lane == 0) { swsum[wave] = ls; swsq[wave] = ls2; }
  __syncthreads();
  float S = 0.f, S2 = 0.f;
#pragma unroll
  for (int w = 0; w < 8; ++w) { S += swsum[w]; S2 += swsq[w]; }

  const float ntot = (float)(O * 64);
  const float mean = S / ntot;
  float var = (S2 - ntot * mean * mean) / (ntot - 1.0f);
  var = fmaxf(var, 0.0f);
  const float inv = 1.0f / (sqrtf(var) + LN_EPS);

  float* ob = out + (long)b * O * 64;
#pragma unroll
  for (int m = 0; m < MT; ++m) {
    const int mtile = wave * MT + m;
#pragma unroll
    for (int t = 0; t < 4; ++t) {
      const int n = t * 16 + nl;
      if (n < 63) {
#pragma unroll
        for (int r = 0; r < 8; ++r) {
          const int o = mtile * 16 + half * 8 + r;
          ob[o * 64 + n + 1] = leaky((acc[m][t][r] - mean) * inv);
        }
      }
    }
  }
  // slot 0 column: zero before norm -> leaky((0 - mean)/std) for every row o
  const float v0 = leaky((0.0f - mean) * inv);
  for (int o = tid; o < O; o += 256) ob[o * 64] = v0;
}

// ---------------------------------------------------------------------------

extern "C" void kernel_launch(void* const* d_in, const int* in_sizes, int n_in,
                              void* d_out, int out_size, void* d_ws, size_t ws_size,
                              hipStream_t stream)
{
  (void)in_sizes; (void)n_in; (void)out_size; (void)ws_size;

  const float* trees = (const float*)d_in[0];
  const int*   idx   = (const int*)  d_in[1];   // (B,189,1) int32
  const float* W1 = (const float*)d_in[2];  const float* b1 = (const float*)d_in[3];
  const float* W2 = (const float*)d_in[4];  const float* b2 = (const float*)d_in[5];
  const float* W3 = (const float*)d_in[6];  const float* b3 = (const float*)d_in[7];
  const float* W4 = (const float*)d_in[8];  const float* b4 = (const float*)d_in[9];
  const float* cw1 = (const float*)d_in[10]; const float* cb1 = (const float*)d_in[11];
  const float* cw2 = (const float*)d_in[12]; const float* cb2 = (const float*)d_in[13];
  const float* cw3 = (const float*)d_in[14]; const float* cb3 = (const float*)d_in[15];

  float* ws = (float*)d_ws;
  // Workspace layout (floats). t2 region is reused for the small MLP intermediates
  // (buf1..buf3 are dead before t2 is first written by conv2).
  float* t2   = ws;                                        // 4096*256*64 = 67,108,864
  float* buf1 = ws;                                        // 4096*64    =    262,144
  float* buf2 = ws + (long)4096 * 64;                      // 4096*256   =  1,048,576
  float* buf3 = ws + (long)4096 * 64 + (long)4096 * 256;   // 4096*1024  =  4,194,304
  float* buf4 = ws + (long)67108864;                       // 4096*4096  = 16,777,216
  float* t1   = buf4 + (long)16777216;                     // 4096*128*64= 33,554,432

  dim3 blk(256);
  auto nblk = [](int M, int N) { return (unsigned)(((M / 16) * (N / 16) + 7) / 8); };

  // MLP stack (B=4096): 16 -> 64 -> 256 -> 1024 -> 4096, LeakyReLU each
  gemm_leaky_kernel<<<nblk(4096, 64),   blk, 0, stream>>>(trees, W1, b1, buf1, 4096, 64,   16);
  gemm_leaky_kernel<<<nblk(4096, 256),  blk, 0, stream>>>(buf1,  W2, b2, buf2, 4096, 256,  64);
  gemm_leaky_kernel<<<nblk(4096, 1024), blk, 0, stream>>>(buf2,  W3, b3, buf3, 4096, 1024, 256);
  gemm_leaky_kernel<<<nblk(4096, 4096), blk, 0, stream>>>(buf3,  W4, b4, buf4, 4096, 4096, 1024);

  // Tree conv + layernorm + leaky, fused per-sample (buf4 viewed as (B, 64, 64))
  conv_ln_leaky_kernel<64, 128> <<<4096, blk, 0, stream>>>(buf4, idx, cw1, cb1, t1);
  conv_ln_leaky_kernel<128, 256><<<4096, blk, 0, stream>>>(t1,   idx, cw2, cb2, t2);
  conv_ln_leaky_kernel<256, 512><<<4096, blk, 0, stream>>>(t2,   idx, cw3, cb3, (float*)d_out);
}